// ConnectedComponentContentEncoder_53661321396445
// MI455X (gfx1250) — compile-verified
//
#include <hip/hip_runtime.h>
#include <hip/hip_bf16.h>
#include <math.h>

#define GH 30
#define GW 30
#define HWC 900
#define DD 256
#define MAXOBJ 16
#define SENT 900
#define KF 261          // D + 5 real features
#define KP 264          // K padded to multiple of 4
#define LD 268          // LDS row stride (floats): 268 % 64 == 12 -> 16 conflict-free rows

typedef __attribute__((ext_vector_type(2))) float v2f;
typedef __attribute__((ext_vector_type(8))) float v8f;

// ---------------------------------------------------------------------------
// Kernel A: connected-component labeling + object extraction (1 block / batch)
// ---------------------------------------------------------------------------
__global__ __launch_bounds__(1024) void cc_extract_kernel(const int* __restrict__ grid,
                                                          int* __restrict__ meta) {
  __shared__ int g[HWC];
  __shared__ int lab[HWC];
  __shared__ int rmn[HWC], rmx[HWC], cmn[HWC], cmx[HWC];
  __shared__ int roots[MAXOBJ];
  __shared__ int nroots;
  __shared__ int changed;

  const int b = blockIdx.x;
  const int tid = threadIdx.x;
  const int* gg = grid + b * HWC;

  for (int t = tid; t < HWC; t += blockDim.x) {
    int v = gg[t];
    g[t] = v;
    lab[t] = (v > 0) ? t : SENT;
    rmn[t] = GH; rmx[t] = -1; cmn[t] = GW; cmx[t] = -1;
  }
  __syncthreads();

  // Jacobi min-propagation with pointer jumping (monotone -> converges)
  for (int iter = 0; iter < 2048; ++iter) {
    if (tid == 0) changed = 0;
    __syncthreads();
    const int cell = tid;
    const bool active = (cell < HWC) && (g[cell] > 0);
    int newl = SENT, oldl = SENT;
    if (active) {
      const int r = cell / GW, c = cell % GW;
      const int gv = g[cell];
      int m = lab[cell]; oldl = m;
      if (r > 0      && g[cell - GW] == gv) m = min(m, lab[cell - GW]);
      if (r < GH - 1 && g[cell + GW] == gv) m = min(m, lab[cell + GW]);
      if (c > 0      && g[cell - 1]  == gv) m = min(m, lab[cell - 1]);
      if (c < GW - 1 && g[cell + 1]  == gv) m = min(m, lab[cell + 1]);
      if (m < SENT) m = min(m, lab[m]);          // pointer jump
      newl = m;
    }
    __syncthreads();
    if (active && newl != oldl) { lab[cell] = newl; changed = 1; }
    __syncthreads();
    const int done = (changed == 0);
    __syncthreads();
    if (done) break;
  }

  // per-component bounding boxes via LDS atomics
  if (tid < HWC && g[tid] > 0) {
    const int root = lab[tid];
    atomicMin(&rmn[root], tid / GW);
    atomicMax(&rmx[root], tid / GW);
    atomicMin(&cmn[root], tid % GW);
    atomicMax(&cmx[root], tid % GW);
  }
  __syncthreads();

  // first 16 roots in ascending linear-index order (matches jnp.sort(root))
  if (tid == 0) {
    int cnt = 0;
    for (int i = 0; i < HWC && cnt < MAXOBJ; ++i)
      if (g[i] > 0 && lab[i] == i) roots[cnt++] = i;
    nroots = cnt;
  }
  __syncthreads();

  if (tid < MAXOBJ) {
    int* mo = meta + (b * MAXOBJ + tid) * 6;
    if (tid < nroots) {
      const int r = roots[tid];
      int y  = min(max(rmn[r], 0), GH - 1);
      int x  = min(max(cmn[r], 0), GW - 1);
      int y2 = min(max(rmx[r] + 1, y + 1), GH);
      int x2 = min(max(cmx[r] + 1, x + 1), GW);
      mo[0] = g[r]; mo[1] = y; mo[2] = x; mo[3] = y2 - y; mo[4] = x2 - x; mo[5] = 1;
    } else {
      // garbage-equivalent slot: values are killed by valid-mask downstream
      mo[0] = 0; mo[1] = 0; mo[2] = 0; mo[3] = 1; mo[4] = 1; mo[5] = 0;
    }
  }
}

// ---------------------------------------------------------------------------
// Kernel B: direct bbox mean-pool (no SAT) + bbox features -> combined[B,16,264]
// ---------------------------------------------------------------------------
__global__ __launch_bounds__(256) void pool_kernel(const float* __restrict__ grid_emb,
                                                   const int* __restrict__ meta,
                                                   float* __restrict__ comb) {
  const int bk = blockIdx.x;            // b*16 + k
  const int b = bk >> 4;
  const int d = threadIdx.x;            // channel, coalesced across the wave
  const int* mo = meta + bk * 6;
  const int color = mo[0], y = mo[1], x = mo[2], h = mo[3], w = mo[4];

  float s = 0.f;
  for (int i = 0; i < h; ++i) {
    const float* rowp = grid_emb + (((size_t)(b * GH + y + i)) * GW + x) * DD + d;
    for (int j = 0; j < w; ++j) s += rowp[(size_t)j * DD];
  }
  float* op = comb + (size_t)bk * KP;
  op[d] = s / (float)(h * w);
  if (d == 0) {
    op[DD + 0] = (float)color / 9.0f;
    op[DD + 1] = (float)x / (float)GW;
    op[DD + 2] = (float)y / (float)GH;
    op[DD + 3] = (float)w / (float)GW;
    op[DD + 4] = (float)h / (float)GH;
    op[DD + 5] = 0.f; op[DD + 6] = 0.f; op[DD + 7] = 0.f;   // K-pad
  }
}

// ---------------------------------------------------------------------------
// Kernel C: fused  gelu(X@W1+b1)@W2+b2 -> mask -> ortho -> @Wp+bp -> LayerNorm
// one block per batch; M=16 rows; 16 waves, each owns one 16-col WMMA tile
// ---------------------------------------------------------------------------
__global__ __launch_bounds__(512) void fused_mlp_kernel(
    const float* __restrict__ comb_g, const int* __restrict__ meta,
    const float* __restrict__ structure, const float* __restrict__ W1,
    const float* __restrict__ b1, const float* __restrict__ W2,
    const float* __restrict__ b2, const float* __restrict__ Wp,
    const float* __restrict__ bp, const float* __restrict__ gmm,
    const float* __restrict__ bta, const float* __restrict__ osp,
    float* __restrict__ out) {
  __shared__ float sA[MAXOBJ][LD];
  __shared__ float sB[MAXOBJ][LD];
  __shared__ float sC[MAXOBJ][LD];
  __shared__ float s_sn[DD];
  __shared__ float s_dot[MAXOBJ];
  __shared__ float s_red[16];
  __shared__ int   s_valid[MAXOBJ];
  __shared__ float s_norm;

  const int b = blockIdx.x;
  const int tid = threadIdx.x;
  const int lane = tid & 31;
  const int wv = tid >> 5;                 // 0..15 : N-tile / row owner
  const int half = lane >> 4;              // f32 WMMA fragment half
  const int mrow = lane & 15;              // A row / B col within tile
  const int nn = wv * 16 + (lane & 15);    // global output column for this lane

  // load combined tile, zero LDS pad columns
  for (int t = tid; t < MAXOBJ * KP; t += 512) {
    int r = t / KP, c = t % KP;
    sC[r][c] = comb_g[(size_t)(b * MAXOBJ + r) * KP + c];
  }
  if (tid < MAXOBJ * (LD - KP)) {
    int r = tid / (LD - KP), c = KP + tid % (LD - KP);
    sC[r][c] = 0.f;
  }
  if (tid < MAXOBJ) s_valid[tid] = meta[(b * MAXOBJ + tid) * 6 + 5];

  // per-batch normalized mean structure vector
  if (tid < DD) {
    const float* sp = structure + (size_t)b * 8 * DD + tid;
    float s = 0.f;
    #pragma unroll
    for (int k = 0; k < 8; ++k) s += sp[(size_t)k * DD];
    s_sn[tid] = s * 0.125f;
  }
  __syncthreads();
  if (tid < DD) {
    float p = s_sn[tid] * s_sn[tid];
    #pragma unroll
    for (int m = 16; m >= 1; m >>= 1) p += __shfl_xor(p, m, 32);
    if (lane == 0) s_red[wv] = p;
  }
  __syncthreads();
  if (tid == 0) {
    float t = 0.f;
    for (int i = 0; i < 8; ++i) t += s_red[i];
    s_norm = fmaxf(sqrtf(t), 1e-8f);
  }
  __syncthreads();
  if (tid < DD) s_sn[tid] /= s_norm;

  // ---- stage 1: hdn = gelu(comb @ W1 + b1), K = 261 (padded 264) ----
  {
    v8f acc = {};
    for (int k0 = 0; k0 < KP; k0 += 4) {
      const int kk = k0 + half * 2;
      v2f av; av.x = sC[mrow][kk]; av.y = sC[mrow][kk + 1];   // zero in pad region
      const int r0 = min(kk, KF - 1), r1 = min(kk + 1, KF - 1); // clamp: A==0 there
      v2f bv; bv.x = W1[(size_t)r0 * DD + nn]; bv.y = W1[(size_t)r1 * DD + nn];
      acc = __builtin_amdgcn_wmma_f32_16x16x4_f32(false, av, false, bv,
                                                  (short)0, acc, false, false);
    }
    const float bias = b1[nn];
    #pragma unroll
    for (int i = 0; i < 8; ++i) {
      const int row = half * 8 + i;
      float v = acc[i] + bias;
      v = 0.5f * v * (1.0f + erff(v * 0.70710678118654752f));   // exact gelu
      sA[row][nn] = v;
    }
  }
  __syncthreads();

  // ---- stage 2: obj = hdn @ W2 + b2, masked by valid ----
  {
    v8f acc = {};
    for (int k0 = 0; k0 < DD; k0 += 4) {
      const int kk = k0 + half * 2;
      v2f av; av.x = sA[mrow][kk]; av.y = sA[mrow][kk + 1];
      v2f bv; bv.x = W2[(size_t)kk * DD + nn]; bv.y = W2[(size_t)(kk + 1) * DD + nn];
      acc = __builtin_amdgcn_wmma_f32_16x16x4_f32(false, av, false, bv,
                                                  (short)0, acc, false, false);
    }
    const float bias = b2[nn];
    #pragma unroll
    for (int i = 0; i < 8; ++i) {
      const int row = half * 8 + i;
      const float v = acc[i] + bias;
      sB[row][nn] = s_valid[row] ? v : 0.f;
    }
  }
  __syncthreads();

  // ---- ortho: co = (obj - (obj.sn) sn) * ortho_scale ----
  {
    float p = 0.f;
    #pragma unroll
    for (int j = 0; j < 8; ++j) {
      const int n = lane + 32 * j;
      p += sB[wv][n] * s_sn[n];
    }
    #pragma unroll
    for (int m = 16; m >= 1; m >>= 1) p += __shfl_xor(p, m, 32);
    if (lane == 0) s_dot[wv] = p;
  }
  __syncthreads();
  {
    const float os = osp[0];
    for (int t = tid; t < MAXOBJ * DD; t += 512) {
      const int r = t >> 8, n = t & 255;
      sB[r][n] = (sB[r][n] - s_dot[r] * s_sn[n]) * os;
    }
  }
  __syncthreads();

  // ---- stage 3: pre = co @ Wp + bp ----
  {
    v8f acc = {};
    for (int k0 = 0; k0 < DD; k0 += 4) {
      const int kk = k0 + half * 2;
      v2f av; av.x = sB[mrow][kk]; av.y = sB[mrow][kk + 1];
      v2f bv; bv.x = Wp[(size_t)kk * DD + nn]; bv.y = Wp[(size_t)(kk + 1) * DD + nn];
      acc = __builtin_amdgcn_wmma_f32_16x16x4_f32(false, av, false, bv,
                                                  (short)0, acc, false, false);
    }
    const float bias = bp[nn];
    #pragma unroll
    for (int i = 0; i < 8; ++i) sC[half * 8 + i][nn] = acc[i] + bias;
  }
  __syncthreads();

  // ---- LayerNorm per row (wave wv owns row wv) ----
  {
    float s1 = 0.f, s2 = 0.f;
    #pragma unroll
    for (int j = 0; j < 8; ++j) {
      const float v = sC[wv][lane + 32 * j];
      s1 += v; s2 += v * v;
    }
    #pragma unroll
    for (int m = 16; m >= 1; m >>= 1) {
      s1 += __shfl_xor(s1, m, 32);
      s2 += __shfl_xor(s2, m, 32);
    }
    const float mu = s1 * (1.0f / 256.0f);
    const float var = s2 * (1.0f / 256.0f) - mu * mu;
    const float rstd = rsqrtf(var + 1e-5f);
    float* op = out + (size_t)(b * MAXOBJ + wv) * DD;
    #pragma unroll
    for (int j = 0; j < 8; ++j) {
      const int n = lane + 32 * j;
      const float v = sC[wv][n];
      op[n] = (v - mu) * rstd * gmm[n] + bta[n];
    }
  }
}

// ---------------------------------------------------------------------------
extern "C" void kernel_launch(void* const* d_in, const int* in_sizes, int n_in,
                              void* d_out, int out_size, void* d_ws, size_t ws_size,
                              hipStream_t stream) {
  const float* grid_emb  = (const float*)d_in[0];
  const int*   grid      = (const int*)  d_in[1];
  const float* structure = (const float*)d_in[2];
  const float* W1        = (const float*)d_in[3];
  const float* b1        = (const float*)d_in[4];
  const float* W2        = (const float*)d_in[5];
  const float* b2        = (const float*)d_in[6];
  const float* Wp        = (const float*)d_in[7];
  const float* bp        = (const float*)d_in[8];
  const float* gmm       = (const float*)d_in[9];
  const float* bta       = (const float*)d_in[10];
  const float* osp       = (const float*)d_in[11];
  float* out = (float*)d_out;

  const int B = in_sizes[1] / HWC;   // 128

  float* comb = (float*)d_ws;                                     // B*16*264 floats
  int*   meta = (int*)((char*)d_ws + (size_t)B * MAXOBJ * KP * sizeof(float)); // B*16*6 ints

  cc_extract_kernel<<<B, 1024, 0, stream>>>(grid, meta);
  pool_kernel<<<B * MAXOBJ, 256, 0, stream>>>(grid_emb, meta, comb);
  fused_mlp_kernel<<<B, 512, 0, stream>>>(comb, meta, structure, W1, b1, W2, b2,
                                          Wp, bp, gmm, bta, osp, out);
}